// LASD_53180285059887
// MI455X (gfx1250) — compile-verified
//
#include <hip/hip_runtime.h>
#include <hip/hip_bf16.h>
#include <stdint.h>

// LASD focal loss, two-pass, bandwidth-bound design for MI455X (gfx1250).
// B=8, C=21, H=W=512. HW = 2^18. NPIX = 2^21. TOTAL = 21 * 2^21.

#define LASD_C      21
#define LASD_HW     (1 << 18)
#define LASD_NPIX   (1 << 21)
#define LASD_TOTAL  (21.0 * 2097152.0)
#define LASD_ALPHA  0.25f
#define LASD_EPS    1e-8f
#define LASD_OHEPS  1e-6f

struct LasdWs {
    unsigned min_bits;   // non-negative float bit pattern -> uint compare == float compare
    unsigned max_bits;
    double   sum;        // global loss accumulator (global_atomic_add_f64)
};

__global__ void lasd_init_ws(LasdWs* ws) {
    ws->min_bits = 0x7f800000u;   // +inf
    ws->max_bits = 0u;            // 0.0f (sigma >= 0)
    ws->sum      = 0.0;
}

// ---------------------------------------------------------------------------
// Pass 1: softmax std per pixel (online max-rescale, no per-class storage),
// write raw sigma, block-reduce min/max, atomics into ws.
// 4 pixels/thread with b128 loads: 21 coalesced global_load_b128 per thread.
// Kept low-VGPR (online) so each wave keeps many HBM loads in flight.
// ---------------------------------------------------------------------------
__global__ __launch_bounds__(256) void lasd_pass1(const float* __restrict__ logits,
                                                  float* __restrict__ sigma_out,
                                                  LasdWs* __restrict__ ws) {
    const int tid  = threadIdx.x;
    const int g    = blockIdx.x * 256 + tid;      // 0 .. 524287
    const int pix0 = g << 2;                      // 4 pixels per thread, same batch b
    const int b    = pix0 >> 18;
    const int hw0  = pix0 & (LASD_HW - 1);
    const float* base = logits + (size_t)b * (size_t)(LASD_C * LASD_HW) + hw0;

    float m[4], s1[4], s2[4];
#pragma unroll
    for (int j = 0; j < 4; ++j) { m[j] = -__builtin_inff(); s1[j] = 0.f; s2[j] = 0.f; }

#pragma unroll
    for (int c = 0; c < LASD_C; ++c) {
        float4 x = *reinterpret_cast<const float4*>(base + (size_t)c * LASD_HW);
        float xv[4] = {x.x, x.y, x.z, x.w};
#pragma unroll
        for (int j = 0; j < 4; ++j) {
            float nm = fmaxf(m[j], xv[j]);
            float sc = __expf(m[j] - nm);         // rescale old accumulators
            float e  = __expf(xv[j] - nm);
            s1[j] = s1[j] * sc + e;
            s2[j] = s2[j] * (sc * sc) + e * e;
            m[j] = nm;
        }
    }

    // sigma = sqrt((sum(p^2) - 1/21) / 20); the +EPS shift cancels in ddof=1 std.
    float vmin = __builtin_inff(), vmax = -__builtin_inff();
    float sg[4];
#pragma unroll
    for (int j = 0; j < 4; ++j) {
        float inv   = 1.0f / s1[j];
        float sump2 = s2[j] * inv * inv;
        float var   = (sump2 - (1.0f / 21.0f)) * (1.0f / 20.0f);
        sg[j] = sqrtf(fmaxf(var, 0.0f));
        vmin = fminf(vmin, sg[j]);
        vmax = fmaxf(vmax, sg[j]);
    }
    // sigma_out = d_out+1 is only 4B-aligned -> scalar b32 stores (still cacheline-dense).
#pragma unroll
    for (int j = 0; j < 4; ++j) sigma_out[pix0 + j] = sg[j];

    // wave32 reduction, then per-block, then global atomics (uint compare valid: sigma >= 0)
#pragma unroll
    for (int off = 16; off; off >>= 1) {
        vmin = fminf(vmin, __shfl_xor(vmin, off, 32));
        vmax = fmaxf(vmax, __shfl_xor(vmax, off, 32));
    }
    __shared__ float wmn[8], wmx[8];
    const int lane = tid & 31, wv = tid >> 5;
    if (lane == 0) { wmn[wv] = vmin; wmx[wv] = vmax; }
    __syncthreads();
    if (tid == 0) {
        float bmn = wmn[0], bmx = wmx[0];
#pragma unroll
        for (int w = 1; w < 8; ++w) { bmn = fminf(bmn, wmn[w]); bmx = fmaxf(bmx, wmx[w]); }
        atomicMin(&ws->min_bits, __float_as_uint(bmn));
        atomicMax(&ws->max_bits, __float_as_uint(bmx));
    }
}

// ---------------------------------------------------------------------------
// Pass 2: double-buffered async global->LDS staging of 256-pixel logit tiles,
// normalize sigma, focal loss, fp64 atomic accumulation.
// Tile = 256 pixels x 21 classes = 21504 B; staged as [class][pixel] words.
// 1344 16B-chunks per tile -> 6 global_load_async_to_lds_b128 per thread.
// Per pixel the 21 staged values are pulled into VGPRs once (single LDS sweep)
// and exp() results are reused for the focal sweep: 21 ds_load + 21 exp + 21 log.
// ---------------------------------------------------------------------------
#define STAGE_WORDS 5376            // 21*256
#define TILE_CHUNKS 1344            // 21*64
#define TILES_PER_BLOCK 4

__device__ __forceinline__ void lasd_issue_tile(const float* __restrict__ logits,
                                                unsigned lds_base, int tile, int buf, int tid) {
    const int pixbase = tile << 8;
    const int b   = pixbase >> 18;
    const int hwb = pixbase & (LASD_HW - 1);
    const float* gbase = logits + (size_t)b * (size_t)(LASD_C * LASD_HW) + hwb;
#pragma unroll
    for (int i = 0; i < 6; ++i) {
        int k       = i * 256 + tid;                       // chunk id (may exceed 1343)
        int valid   = (k < TILE_CHUNKS);
        int kk      = valid ? k : (TILE_CHUNKS - 1);
        int c       = kk >> 6;
        int g4      = (kk & 63) << 2;                      // pixel offset within tile
        unsigned ldsoff = valid ? (lds_base + (unsigned)(buf * STAGE_WORDS + kk * 4) * 4u)
                                : (lds_base + (unsigned)(2 * STAGE_WORDS) * 4u); // pad slot
        unsigned long long ga =
            (unsigned long long)(uintptr_t)(gbase + (size_t)c * LASD_HW + g4);
        asm volatile("global_load_async_to_lds_b128 %0, %1, off"
                     :: "v"(ldsoff), "v"(ga) : "memory");
    }
}

__global__ __launch_bounds__(256) void lasd_pass2(const float* __restrict__ logits,
                                                  const int* __restrict__ target,
                                                  float* __restrict__ sigma_io,
                                                  LasdWs* __restrict__ ws) {
    __shared__ float stage[2 * STAGE_WORDS + 4];   // 2 buffers + 16B dummy pad
    const unsigned lds_base = (unsigned)(uintptr_t)&stage[0];
    const int tid = threadIdx.x;

    const float smin = __uint_as_float(ws->min_bits);
    const float smax = __uint_as_float(ws->max_bits);
    const float inv_range = 1.0f / (smax - smin);

    const int tile0 = blockIdx.x * TILES_PER_BLOCK;
    lasd_issue_tile(logits, lds_base, tile0, /*buf=*/0, tid);

    float acc = 0.0f;
    for (int i = 0; i < TILES_PER_BLOCK; ++i) {
        const int buf = i & 1;
        // Wait for this wave's async loads (tile i and earlier), then barrier so
        // every wave's chunks are visible, and every wave is done reading buf^1
        // before it is overwritten with tile i+1.
        asm volatile("s_wait_asynccnt 0" ::: "memory");
        __syncthreads();
        if (i + 1 < TILES_PER_BLOCK)
            lasd_issue_tile(logits, lds_base, tile0 + i + 1, buf ^ 1, tid);

        const int n = (tile0 + i) * 256 + tid;
        const float* xs = &stage[buf * STAGE_WORDS];

        // Single LDS sweep into VGPRs + row max.
        float e[LASD_C];
        float mmax = -__builtin_inff();
#pragma unroll
        for (int c = 0; c < LASD_C; ++c) {
            e[c] = xs[c * 256 + tid];
            mmax = fmaxf(mmax, e[c]);
        }
        // exp once, reuse for both the denominator and the focal term.
        float s1 = 0.0f;
#pragma unroll
        for (int c = 0; c < LASD_C; ++c) {
            e[c] = __expf(e[c] - mmax);
            s1 += e[c];
        }
        const float invs = 1.0f / s1;

        float sr = sigma_io[n];
        float sn = (sr - smin) * inv_range;
        sigma_io[n] = sn;                          // normalized sigma is the 2nd output
        const float beta = __expf(-sn) + 1.0f;
        const int   tg   = target[n];

        float a = 0.0f;
#pragma unroll
        for (int c = 0; c < LASD_C; ++c) {
            float p   = e[c] * invs + LASD_EPS;
            float omp = 1.0f - p;
            float f   = omp * omp * __logf(p);     // sign/ALPHA/beta folded below
            float oh  = (c == tg) ? (1.0f + LASD_OHEPS) : LASD_OHEPS;
            a += oh * f;
        }
        acc += a * (-LASD_ALPHA) * beta;
    }

    // block reduction -> one fp64 atomic per block
#pragma unroll
    for (int off = 16; off; off >>= 1) acc += __shfl_xor(acc, off, 32);
    __shared__ float wsum[8];
    const int lane = tid & 31, wv = tid >> 5;
    if (lane == 0) wsum[wv] = acc;
    __syncthreads();
    if (tid == 0) {
        float t = 0.0f;
#pragma unroll
        for (int w = 0; w < 8; ++w) t += wsum[w];
        atomicAdd(&ws->sum, (double)t);
    }
}

__global__ void lasd_finalize(const LasdWs* __restrict__ ws, float* __restrict__ out) {
    out[0] = (float)(ws->sum / LASD_TOTAL);
}

extern "C" void kernel_launch(void* const* d_in, const int* in_sizes, int n_in,
                              void* d_out, int out_size, void* d_ws, size_t ws_size,
                              hipStream_t stream) {
    (void)in_sizes; (void)n_in; (void)out_size; (void)ws_size;
    const float* logits = (const float*)d_in[0];
    const int*   target = (const int*)d_in[1];
    float*       out    = (float*)d_out;
    float*       sigma  = out + 1;                 // (loss, sigma) concatenated
    LasdWs*      ws     = (LasdWs*)d_ws;

    lasd_init_ws<<<1, 1, 0, stream>>>(ws);
    lasd_pass1<<<LASD_NPIX / (256 * 4), 256, 0, stream>>>(logits, sigma, ws);
    lasd_pass2<<<(LASD_NPIX / 256) / TILES_PER_BLOCK, 256, 0, stream>>>(logits, target, sigma, ws);
    lasd_finalize<<<1, 1, 0, stream>>>(ws, out);
}